// TatumPooling_66322884984856
// MI455X (gfx1250) — compile-verified
//
#include <hip/hip_runtime.h>
#include <math.h>

// Problem constants from the reference: B=32, F=16384, C=96, T=1024
#define TP_B 32
#define TP_F 16384
#define TP_C 96
#define TP_T 1024
#define SEGS (TP_B * TP_T)           // 32768 segments
#define WAVES_PER_BLOCK 8            // 256 threads = 8 wave32

typedef __attribute__((ext_vector_type(4))) float fvec4;

// One wave32 per (b, t) segment. Lanes 0..23 each own 4 channels (float4).
// Per frame: one coalesced 384B global_load_b128 (non-temporal) per wave.
__global__ __launch_bounds__(256)
void tatum_pool_kernel(const float* __restrict__ x,
                       const long long* __restrict__ bounds,
                       float* __restrict__ out) {
    // Wave id within block, made wave-uniform (SGPR) so the frame loop is
    // scalar-controlled (s_cmp/s_cbranch) instead of an exec-mask loop.
    const int wave = __builtin_amdgcn_readfirstlane((int)threadIdx.x) >> 5;
    const int lane = (int)threadIdx.x & 31;

    const int seg = (int)blockIdx.x * WAVES_PER_BLOCK + wave;  // 0 .. SEGS-1
    const int b   = seg >> 10;                                  // seg / T

    // Boundaries: int64 pairs, uniform per wave -> scalar loads.
    const int start = (int)bounds[2 * (long long)seg];
    const int stop  = (int)bounds[2 * (long long)seg + 1];

    if (lane < (TP_C / 4)) {  // 24 active lanes cover 96 channels as float4
        const fvec4* base = (const fvec4*)(x + (size_t)b * TP_F * TP_C);
        fvec4 m = { -INFINITY, -INFINITY, -INFINITY, -INFINITY };

        for (int f = start; f < stop; ++f) {
            const fvec4* p = base + (size_t)f * (TP_C / 4) + lane;
            // Speculative prefetch ~2 frames (768B) ahead of the stream;
            // lowers to global_prefetch_b8, dropped silently if OOB.
            __builtin_prefetch((const void*)(p + 2 * (TP_C / 4)), 0, 0);
            // Read-once stream: non-temporal load keeps L2 clean.
            fvec4 v = __builtin_nontemporal_load(p);
            m.x = fmaxf(m.x, v.x);
            m.y = fmaxf(m.y, v.y);
            m.z = fmaxf(m.z, v.z);
            m.w = fmaxf(m.w, v.w);
        }

        fvec4* o = (fvec4*)out + (size_t)seg * (TP_C / 4) + lane;
        __builtin_nontemporal_store(m, o);
    }
}

extern "C" void kernel_launch(void* const* d_in, const int* in_sizes, int n_in,
                              void* d_out, int out_size, void* d_ws, size_t ws_size,
                              hipStream_t stream) {
    (void)in_sizes; (void)n_in; (void)out_size; (void)d_ws; (void)ws_size;
    const float*     x      = (const float*)d_in[0];      // (B, F, C) float32
    const long long* bounds = (const long long*)d_in[1];  // (B, T, 2) int64
    float*           out    = (float*)d_out;              // (B, T, C) float32

    const int blocks = SEGS / WAVES_PER_BLOCK;  // 4096
    tatum_pool_kernel<<<blocks, 256, 0, stream>>>(x, bounds, out);
}